// MoEFeedForward_19731079758428
// MI455X (gfx1250) — compile-verified
//
#include <hip/hip_runtime.h>
#include <math.h>

// ---- Problem constants (match reference) ----
#define E_  8        // experts
#define H_  1024     // hidden
#define I_  4096     // inner
#define T_  8192     // tokens (B*S = 4*2048)
#define C_  1280     // capacity = ceil(1.25*8192/8)
#define LOG2T 13     // T_ == 1<<13

#define BK_   32     // K-chunk staged in LDS
#define APITCH 36    // LDS row pitch (floats): 144B -> 16B aligned, bank-conflict free

typedef __attribute__((ext_vector_type(2))) float v2f;
typedef __attribute__((ext_vector_type(4))) float v4f;
typedef __attribute__((ext_vector_type(8))) float v8f;

// ---------------------------------------------------------------------------
// 0) zero-fill (capacity buffer must start at 0: under-filled expert slots)
// ---------------------------------------------------------------------------
__global__ void moe_zero_kernel(float* __restrict__ p, size_t n) {
    size_t i = (size_t)blockIdx.x * blockDim.x + threadIdx.x;
    size_t stride = (size_t)gridDim.x * blockDim.x;
    for (; i < n; i += stride) p[i] = 0.0f;
}

// ---------------------------------------------------------------------------
// 1) Router: logits = x @ w_router ; softmax ; top-2 ; logsumexp^2
//    w_router (1024x8 = 32KB) staged in LDS (320KB per WGP available).
// ---------------------------------------------------------------------------
__global__ __launch_bounds__(256)
void moe_router_kernel(const float* __restrict__ x, const float* __restrict__ wr,
                       float* __restrict__ probs, float* __restrict__ zterm,
                       int* __restrict__ top_idx, float* __restrict__ top_val) {
    __shared__ float sw[H_ * E_];
    for (int i = threadIdx.x; i < H_ * E_; i += blockDim.x) sw[i] = wr[i];
    __syncthreads();

    const int t = blockIdx.x * blockDim.x + threadIdx.x;
    const float* xr = x + (size_t)t * H_;

    float acc[E_];
#pragma unroll
    for (int e = 0; e < E_; ++e) acc[e] = 0.0f;
    for (int hh = 0; hh < H_; ++hh) {
        float xv = xr[hh];
#pragma unroll
        for (int e = 0; e < E_; ++e) acc[e] += xv * sw[hh * E_ + e];
    }
    float mx = acc[0];
#pragma unroll
    for (int e = 1; e < E_; ++e) mx = fmaxf(mx, acc[e]);
    float p[E_], s = 0.0f;
#pragma unroll
    for (int e = 0; e < E_; ++e) { p[e] = __expf(acc[e] - mx); s += p[e]; }
    float inv = 1.0f / s;
#pragma unroll
    for (int e = 0; e < E_; ++e) { p[e] *= inv; probs[t * E_ + e] = p[e]; }
    float lse = mx + __logf(s);
    zterm[t] = lse * lse;

    // top-2 (strict > keeps lowest index on ties, matching jax top_k)
    int i0 = 0; float v0 = p[0];
#pragma unroll
    for (int e = 1; e < E_; ++e) if (p[e] > v0) { v0 = p[e]; i0 = e; }
    int i1 = (i0 == 0) ? 1 : 0; float v1 = p[i1];
#pragma unroll
    for (int e = 0; e < E_; ++e)
        if (e != i0 && p[e] > v1) { v1 = p[e]; i1 = e; }

    top_idx[t * 2 + 0] = i0; top_idx[t * 2 + 1] = i1;
    top_val[t * 2 + 0] = v0; top_val[t * 2 + 1] = v1;
}

// ---------------------------------------------------------------------------
// 2) Ordered capacity scan. Reference order: j = k*T + t over 2T entries.
//    One block, 8 waves; wave e scans its expert with ballot+prefix-popcount.
//    Deterministic (no atomics) and wave32-native.
// ---------------------------------------------------------------------------
__global__ __launch_bounds__(256)
void moe_scan_kernel(const int* __restrict__ top_idx, int* __restrict__ pos_arr,
                     int* __restrict__ cnt) {
    const int e    = threadIdx.x >> 5;   // wave id == expert id
    const int lane = threadIdx.x & 31;
    int base = 0;
    for (int j0 = 0; j0 < 2 * T_; j0 += 32) {
        int j = j0 + lane;
        int t = j & (T_ - 1);
        int k = j >> LOG2T;
        int ex = top_idx[t * 2 + k];
        bool match = (ex == e);
        unsigned mask = __builtin_amdgcn_ballot_w32(match);
        int prefix = __popc(mask & ((1u << lane) - 1u));
        if (match) pos_arr[j] = base + prefix;
        base += __popc(mask);
    }
    if (lane == 0) cnt[e] = (base < C_) ? base : C_;   // kept count per expert
}

// ---------------------------------------------------------------------------
// 3) Scatter tokens into capacity buffer buf[e][pos][H] (dropped if pos>=C)
// ---------------------------------------------------------------------------
__global__ __launch_bounds__(256)
void moe_scatter_kernel(const float* __restrict__ x, const int* __restrict__ top_idx,
                        const int* __restrict__ pos_arr, float* __restrict__ buf) {
    const int j = blockIdx.x;
    const int t = j & (T_ - 1);
    const int k = j >> LOG2T;
    const int e = top_idx[t * 2 + k];
    const int p = pos_arr[j];
    if (p >= C_) return;
    const v4f* src = (const v4f*)(x + (size_t)t * H_);
    v4f*       dst = (v4f*)(buf + ((size_t)e * C_ + p) * H_);
    for (int i = threadIdx.x; i < H_ / 4; i += blockDim.x) dst[i] = src[i];
}

// ---------------------------------------------------------------------------
// 4) Fused gate+up WMMA GEMM with double-buffered LDS staging of the A tile.
//    g = buf@Wg + bg ; u = buf@Wu + bu ; h = silu(g)*u
//    f32 tensor path: V_WMMA_F32_16X16X4_F32.
//    Block = 8 waves = 128 output rows; wave owns 16x64 strip.
//    A tile 128xBK staged cooperatively (coalesced b128) into padded LDS
//    (pitch 36 floats: 36*m mod 64 injective over m=0..15 -> conflict-free
//    ds_load_b64 fragment reads; 144B pitch keeps 16B alignment).
//    B stays in global: 64B/half-wave coalesced, L2-resident (32MB/expert vs
//    192MB L2); next-chunk rows prefetched via global_prefetch_b8.
// ---------------------------------------------------------------------------
__global__ __launch_bounds__(256)
void moe_gateup_kernel(const float* __restrict__ buf,
                       const float* __restrict__ wg, const float* __restrict__ bg,
                       const float* __restrict__ wu, const float* __restrict__ bu,
                       float* __restrict__ hout) {
    __shared__ float sA[2][128 * APITCH];   // 2 x 18KB

    const int e    = blockIdx.z;
    const int wave = threadIdx.x >> 5;
    const int lane = threadIdx.x & 31;
    const int half = lane >> 4;
    const int m    = lane & 15;
    const int r0   = blockIdx.y * 128 + wave * 16;  // output row base (this wave)
    const int c0   = blockIdx.x * 64;               // output col base

    const float* Ablk = buf + ((size_t)e * C_ + blockIdx.y * 128) * H_;
    const float* WG = wg + (size_t)e * H_ * I_;
    const float* WU = wu + (size_t)e * H_ * I_;

    // cooperative-staging coordinates: 128 rows x 8 v4f groups = 1024 tasks
    const int srow = threadIdx.x >> 3;      // 0..31 (+32*q)
    const int sg   = threadIdx.x & 7;       // v4f group within BK

    v8f accg[4] = {{0,0,0,0,0,0,0,0},{0,0,0,0,0,0,0,0},{0,0,0,0,0,0,0,0},{0,0,0,0,0,0,0,0}};
    v8f accu[4] = {{0,0,0,0,0,0,0,0},{0,0,0,0,0,0,0,0},{0,0,0,0,0,0,0,0},{0,0,0,0,0,0,0,0}};

    v4f stage[4];
#pragma unroll
    for (int q = 0; q < 4; ++q)
        stage[q] = *(const v4f*)(Ablk + (size_t)(srow + q * 32) * H_ + sg * 4);
#pragma unroll
    for (int q = 0; q < 4; ++q)
        *(v4f*)&sA[0][(srow + q * 32) * APITCH + sg * 4] = stage[q];
    __syncthreads();

    const int nch = H_ / BK_;
    for (int ch = 0; ch < nch; ++ch) {
        const int cur = ch & 1;
        const int k0  = ch * BK_;

        if (ch + 1 < nch) {
            // issue next chunk's A-tile global loads early (overlap with WMMA)
#pragma unroll
            for (int q = 0; q < 4; ++q)
                stage[q] = *(const v4f*)(Ablk + (size_t)(srow + q * 32) * H_ + (k0 + BK_) + sg * 4);
            // hint next chunk's B rows into cache (global_prefetch_b8)
            const int rbn = k0 + BK_ + 2 * half;
#pragma unroll
            for (int tt = 0; tt < 4; ++tt) {
                __builtin_prefetch(&WG[(size_t)rbn * I_ + c0 + tt * 16 + m], 0, 1);
                __builtin_prefetch(&WU[(size_t)rbn * I_ + c0 + tt * 16 + m], 0, 1);
            }
        }

        const float* As = &sA[cur][(wave * 16 + m) * APITCH];
#pragma unroll
        for (int kk = 0; kk < BK_; kk += 4) {
            const int rb = k0 + kk + 2 * half;
            v2f a = *(const v2f*)(As + kk + 2 * half);   // ds_load_b64, conflict-free
#pragma unroll
            for (int tt = 0; tt < 4; ++tt) {
                const int col = c0 + tt * 16 + m;
                v2f bgf; bgf.x = WG[(size_t)rb * I_ + col];
                         bgf.y = WG[(size_t)(rb + 1) * I_ + col];
                accg[tt] = __builtin_amdgcn_wmma_f32_16x16x4_f32(
                    false, a, false, bgf, (short)0, accg[tt], false, false);
                v2f buf_; buf_.x = WU[(size_t)rb * I_ + col];
                          buf_.y = WU[(size_t)(rb + 1) * I_ + col];
                accu[tt] = __builtin_amdgcn_wmma_f32_16x16x4_f32(
                    false, a, false, buf_, (short)0, accu[tt], false, false);
            }
        }

        if (ch + 1 < nch) {
#pragma unroll
            for (int q = 0; q < 4; ++q)
                *(v4f*)&sA[cur ^ 1][(srow + q * 32) * APITCH + sg * 4] = stage[q];
        }
        __syncthreads();
    }

    // epilogue: bias + silu(g)*u, store h
#pragma unroll
    for (int tt = 0; tt < 4; ++tt) {
        const int col = c0 + tt * 16 + m;
        const float bgv = bg[e * I_ + col];
        const float buv = bu[e * I_ + col];
#pragma unroll
        for (int r = 0; r < 8; ++r) {
            float g = accg[tt][r] + bgv;
            float u = accu[tt][r] + buv;
            float hv = (g / (1.0f + __expf(-g))) * u;   // silu(g)*u
            const int row = r0 + half * 8 + r;          // C/D layout: lanes16-31 -> M+8
            hout[((size_t)e * C_ + row) * I_ + col] = hv;
        }
    }
}

// ---------------------------------------------------------------------------
// 5) Down WMMA GEMM: d = h @ Wd + bd   (K = 4096), same staging scheme
// ---------------------------------------------------------------------------
__global__ __launch_bounds__(256)
void moe_down_kernel(const float* __restrict__ hbuf,
                     const float* __restrict__ wd, const float* __restrict__ bd,
                     float* __restrict__ dout) {
    __shared__ float sA[2][128 * APITCH];

    const int e    = blockIdx.z;
    const int wave = threadIdx.x >> 5;
    const int lane = threadIdx.x & 31;
    const int half = lane >> 4;
    const int m    = lane & 15;
    const int r0   = blockIdx.y * 128 + wave * 16;
    const int c0   = blockIdx.x * 64;

    const float* Ablk = hbuf + ((size_t)e * C_ + blockIdx.y * 128) * I_;
    const float* W = wd + (size_t)e * I_ * H_;

    const int srow = threadIdx.x >> 3;
    const int sg   = threadIdx.x & 7;

    v8f acc[4] = {{0,0,0,0,0,0,0,0},{0,0,0,0,0,0,0,0},{0,0,0,0,0,0,0,0},{0,0,0,0,0,0,0,0}};

    v4f stage[4];
#pragma unroll
    for (int q = 0; q < 4; ++q)
        stage[q] = *(const v4f*)(Ablk + (size_t)(srow + q * 32) * I_ + sg * 4);
#pragma unroll
    for (int q = 0; q < 4; ++q)
        *(v4f*)&sA[0][(srow + q * 32) * APITCH + sg * 4] = stage[q];
    __syncthreads();

    const int nch = I_ / BK_;
    for (int ch = 0; ch < nch; ++ch) {
        const int cur = ch & 1;
        const int k0  = ch * BK_;

        if (ch + 1 < nch) {
#pragma unroll
            for (int q = 0; q < 4; ++q)
                stage[q] = *(const v4f*)(Ablk + (size_t)(srow + q * 32) * I_ + (k0 + BK_) + sg * 4);
            const int rbn = k0 + BK_ + 2 * half;
#pragma unroll
            for (int tt = 0; tt < 4; ++tt)
                __builtin_prefetch(&W[(size_t)rbn * H_ + c0 + tt * 16 + m], 0, 1);
        }

        const float* As = &sA[cur][(wave * 16 + m) * APITCH];
#pragma unroll
        for (int kk = 0; kk < BK_; kk += 4) {
            const int rb = k0 + kk + 2 * half;
            v2f a = *(const v2f*)(As + kk + 2 * half);
#pragma unroll
            for (int tt = 0; tt < 4; ++tt) {
                const int col = c0 + tt * 16 + m;
                v2f b; b.x = W[(size_t)rb * H_ + col];
                       b.y = W[(size_t)(rb + 1) * H_ + col];
                acc[tt] = __builtin_amdgcn_wmma_f32_16x16x4_f32(
                    false, a, false, b, (short)0, acc[tt], false, false);
            }
        }

        if (ch + 1 < nch) {
#pragma unroll
            for (int q = 0; q < 4; ++q)
                *(v4f*)&sA[cur ^ 1][(srow + q * 32) * APITCH + sg * 4] = stage[q];
        }
        __syncthreads();
    }

#pragma unroll
    for (int tt = 0; tt < 4; ++tt) {
        const int col = c0 + tt * 16 + m;
        const float bv = bd[e * H_ + col];
#pragma unroll
        for (int r = 0; r < 8; ++r) {
            const int row = r0 + half * 8 + r;
            dout[((size_t)e * C_ + row) * H_ + col] = acc[tt][r] + bv;
        }
    }
}

// ---------------------------------------------------------------------------
// 6) Combine: out[t] = sum_k weight_k * d[e_k][pos_k]  (each token written once
//    -> deterministic, no atomics)
// ---------------------------------------------------------------------------
__global__ __launch_bounds__(256)
void moe_combine_kernel(const float* __restrict__ dbuf, const int* __restrict__ top_idx,
                        const float* __restrict__ top_val, const int* __restrict__ pos_arr,
                        float* __restrict__ out) {
    const int t = blockIdx.x;
    const int e0 = top_idx[t * 2 + 0], e1 = top_idx[t * 2 + 1];
    const int p0 = pos_arr[t];            // j = 0*T + t
    const int p1 = pos_arr[T_ + t];       // j = 1*T + t
    const float w0 = (p0 < C_) ? top_val[t * 2 + 0] : 0.0f;
    const float w1 = (p1 < C_) ? top_val[t * 2 + 1] : 0.0f;
    const int q0 = (p0 < C_) ? p0 : (C_ - 1);
    const int q1 = (p1 < C_) ? p1 : (C_ - 1);
    const v4f* d0 = (const v4f*)(dbuf + ((size_t)e0 * C_ + q0) * H_);
    const v4f* d1 = (const v4f*)(dbuf + ((size_t)e1 * C_ + q1) * H_);
    v4f* o = (v4f*)(out + (size_t)t * H_);
    for (int i = threadIdx.x; i < H_ / 4; i += blockDim.x)
        o[i] = d0[i] * w0 + d1[i] * w1;
}

// ---------------------------------------------------------------------------
// 7) Losses: aux = E * sum_e mean_t(probs[:,e]) * (kept_e/T);  z = mean(lse^2)
//    Single block, fixed-order tree reduction -> deterministic.
// ---------------------------------------------------------------------------
__global__ __launch_bounds__(256)
void moe_loss_kernel(const float* __restrict__ probs, const float* __restrict__ zterm,
                     const int* __restrict__ cnt, float* __restrict__ out_losses) {
    __shared__ float sp[256 * E_];
    __shared__ float sz[256];
    const int tid = threadIdx.x;
    float pa[E_];
#pragma unroll
    for (int e = 0; e < E_; ++e) pa[e] = 0.0f;
    float za = 0.0f;
    for (int t = tid; t < T_; t += 256) {
        za += zterm[t];
#pragma unroll
        for (int e = 0; e < E_; ++e) pa[e] += probs[t * E_ + e];
    }
    sz[tid] = za;
#pragma unroll
    for (int e = 0; e < E_; ++e) sp[tid * E_ + e] = pa[e];
    __syncthreads();
    for (int s = 128; s > 0; s >>= 1) {
        if (tid < s) {
            sz[tid] += sz[tid + s];
#pragma unroll
            for (int e = 0; e < E_; ++e) sp[tid * E_ + e] += sp[(tid + s) * E_ + e];
        }
        __syncthreads();
    }
    if (tid == 0) {
        float aux = 0.0f;
        for (int e = 0; e < E_; ++e)
            aux += (sp[e] / (float)T_) * ((float)cnt[e] / (float)T_);
        out_losses[0] = aux * (float)E_;
        out_losses[1] = sz[0] / (float)T_;
    }
}

// ---------------------------------------------------------------------------
extern "C" void kernel_launch(void* const* d_in, const int* in_sizes, int n_in,
                              void* d_out, int out_size, void* d_ws, size_t ws_size,
                              hipStream_t stream) {
    const float* x  = (const float*)d_in[0];
    const float* wr = (const float*)d_in[1];
    const float* wg = (const float*)d_in[2];
    const float* bg = (const float*)d_in[3];
    const float* wu = (const float*)d_in[4];
    const float* bu = (const float*)d_in[5];
    const float* wd = (const float*)d_in[6];
    const float* bd = (const float*)d_in[7];
    float* out = (float*)d_out;

    // workspace carve-out (~252 MB total)
    char* ws = (char*)d_ws;
    size_t off = 0;
    auto carve = [&](size_t bytes) -> void* {
        void* p = ws + off;
        off += (bytes + 255) & ~(size_t)255;
        return p;
    };
    float* probs   = (float*)carve((size_t)T_ * E_ * 4);
    float* zterm   = (float*)carve((size_t)T_ * 4);
    int*   top_idx = (int*)  carve((size_t)T_ * 2 * 4);
    float* top_val = (float*)carve((size_t)T_ * 2 * 4);
    int*   pos_arr = (int*)  carve((size_t)2 * T_ * 4);
    int*   cnt     = (int*)  carve((size_t)E_ * 4);
    float* buf     = (float*)carve((size_t)E_ * C_ * H_ * 4);   //  41.9 MB
    float* hbuf    = (float*)carve((size_t)E_ * C_ * I_ * 4);   // 167.8 MB
    float* dbuf    = (float*)carve((size_t)E_ * C_ * H_ * 4);   //  41.9 MB
    (void)ws_size; (void)in_sizes; (void)n_in; (void)out_size;

    moe_zero_kernel<<<2048, 256, 0, stream>>>(buf, (size_t)E_ * C_ * H_);
    moe_router_kernel<<<T_ / 256, 256, 0, stream>>>(x, wr, probs, zterm, top_idx, top_val);
    moe_scan_kernel<<<1, 256, 0, stream>>>(top_idx, pos_arr, cnt);
    moe_scatter_kernel<<<2 * T_, 256, 0, stream>>>(x, top_idx, pos_arr, buf);
    moe_gateup_kernel<<<dim3(I_ / 64, C_ / 128, E_), 256, 0, stream>>>(buf, wg, bg, wu, bu, hbuf);
    moe_down_kernel<<<dim3(H_ / 64, C_ / 128, E_), 256, 0, stream>>>(hbuf, wd, bd, dbuf);
    moe_combine_kernel<<<T_, 256, 0, stream>>>(dbuf, top_idx, top_val, pos_arr, out);
    moe_loss_kernel<<<1, 256, 0, stream>>>(probs, zterm, cnt, out + (size_t)T_ * H_);
}